// DecoderV1_18631568130306
// MI455X (gfx1250) — compile-verified
//
#include <hip/hip_runtime.h>
#include <hip/hip_bf16.h>
#include <stdint.h>

// ---------------------------------------------------------------------------
// DecoderV1 (YOLO-style decode + topk + trivial NMS) for MI455X / gfx1250.
//
// keep-mask is identity (IoU never > 1.0, score thr keeps all), so the task is
// an exact per-image top-1000 over 409600 scores + box decode. Bandwidth-bound:
// 52 MB/scan @ 23.3 TB/s. We use the CDNA5 async global->LDS streaming path
// (global_load_async_to_lds_b128 + s_wait_asynccnt) with per-wave double
// buffering for all full-tensor scans.
// ---------------------------------------------------------------------------

#define BATCH     32
#define HW        409600            // 640*640
#define WIDTH     640
#define NBLK      10                // blocks per batch for scan kernels
#define BLK_ELEMS 40960             // HW / NBLK
#define WAVE_ELEMS 5120             // BLK_ELEMS / 8 waves
#define TILE_ELEMS 512              // elements per async tile per wave
#define NTILES    10                // WAVE_ELEMS / TILE_ELEMS
#define K_TOP     1000
#define CAP       2048              // candidate buffer per batch

// ---- order-preserving float -> uint mapping -------------------------------
__device__ __forceinline__ unsigned ordOf(float f) {
  unsigned u = __float_as_uint(f);
  return u ^ ((u & 0x80000000u) ? 0xFFFFFFFFu : 0x80000000u);
}

// ---- async-tensor streaming: issue one 512-element tile (4 x b128/lane) ----
// GVS addressing: 64-bit SGPR base + 32-bit per-lane byte offset. The LDS
// destination address is the low 32 bits of the generic shared pointer.
__device__ __forceinline__ void issue_tile(unsigned long long gbase,
                                           unsigned byteOff,
                                           const float* ldsbuf, int lane) {
  unsigned lbase = (unsigned)(uintptr_t)ldsbuf;
#pragma unroll
  for (int s = 0; s < 4; ++s) {
    unsigned sub   = (unsigned)(s * 512 + lane * 16);
    unsigned voff  = byteOff + sub;
    unsigned laddr = lbase + sub;
    asm volatile("global_load_async_to_lds_b128 %0, %1, %2 offset:0"
                 :: "v"(laddr), "v"(voff), "s"(gbase)
                 : "memory");
  }
}
__device__ __forceinline__ void wait_async_le4() {
  asm volatile("s_wait_asynccnt 0x4" ::: "memory");
}
__device__ __forceinline__ void wait_async_0() {
  asm volatile("s_wait_asynccnt 0x0" ::: "memory");
}

// ---------------------------------------------------------------------------
// init: zero histograms / counters, seed per-batch radix state.
// ---------------------------------------------------------------------------
__global__ void __launch_bounds__(256)
init_kernel(unsigned* __restrict__ ghist, unsigned* __restrict__ state,
            unsigned* __restrict__ gcnt) {
  int t = blockIdx.x * blockDim.x + threadIdx.x;
  if (t < BATCH * 256) ghist[t] = 0u;
  if (t < BATCH) {
    state[t * 2 + 0] = 0u;      // prefix (accumulated high bits of T)
    state[t * 2 + 1] = K_TOP;   // remaining k within current prefix
    gcnt[t] = 0u;
  }
}

// ---------------------------------------------------------------------------
// hist: streaming 256-bin histogram of radix digit `level` (0..3), restricted
// to elements matching the already-selected prefix. Async double-buffered.
// ---------------------------------------------------------------------------
__global__ void __launch_bounds__(256)
hist_kernel(const float* __restrict__ preds, unsigned* __restrict__ ghist,
            const unsigned* __restrict__ state, int level) {
  __shared__ __align__(16) float stage[8][2][TILE_ELEMS];
  __shared__ unsigned lhist[256];
  const int b = blockIdx.y;
  const int tid = threadIdx.x;
  const int wave = tid >> 5, lane = tid & 31;
  lhist[tid] = 0u;
  __syncthreads();

  const unsigned long long gbase =
      (unsigned long long)(preds + (size_t)b * 3 * HW +
                           (size_t)blockIdx.x * BLK_ELEMS);
  const unsigned prefix = (level > 0) ? state[b * 2 + 0] : 0u;
  const int pshift = 32 - 8 * level;   // used only when level > 0
  const int bshift = 24 - 8 * level;
  const unsigned waveByte = (unsigned)(wave * WAVE_ELEMS * 4);

  issue_tile(gbase, waveByte, stage[wave][0], lane);
  int parity = 0;
  for (int t = 0; t < NTILES; ++t) {
    if (t + 1 < NTILES) {
      issue_tile(gbase, waveByte + (unsigned)((t + 1) * TILE_ELEMS * 4),
                 stage[wave][parity ^ 1], lane);
      wait_async_le4();
    } else {
      wait_async_0();
    }
    const float* buf = stage[wave][parity];
#pragma unroll
    for (int s = 0; s < 4; ++s) {
      const float4 v = *(const float4*)(buf + s * 128 + lane * 4);
      float vals[4] = {v.x, v.y, v.z, v.w};
#pragma unroll
      for (int j = 0; j < 4; ++j) {
        unsigned ord = ordOf(vals[j]);
        bool pass = (level == 0) || ((ord >> pshift) == prefix);
        if (pass) atomicAdd(&lhist[(ord >> bshift) & 0xFFu], 1u);
      }
    }
    parity ^= 1;
  }
  __syncthreads();
  atomicAdd(&ghist[b * 256 + tid], lhist[tid]);
}

// ---------------------------------------------------------------------------
// select: per batch, pick the radix bin containing the K-th largest value,
// update (prefix, k_remaining), and zero the histogram for the next level.
// ---------------------------------------------------------------------------
__global__ void __launch_bounds__(256)
select_kernel(unsigned* __restrict__ ghist, unsigned* __restrict__ state) {
  __shared__ unsigned c[256];
  const int b = blockIdx.x, tid = threadIdx.x;
  c[tid] = ghist[b * 256 + tid];
  ghist[b * 256 + tid] = 0u;     // reset for the next radix level
  __syncthreads();
  if (tid == 0) {
    unsigned kcur = state[b * 2 + 1];
    unsigned acc = 0;
    int sel = 0;
    for (int bin = 255; bin >= 0; --bin) {
      unsigned cb = c[bin];
      if (acc + cb >= kcur) { sel = bin; break; }
      acc += cb;
    }
    state[b * 2 + 0] = (state[b * 2 + 0] << 8) | (unsigned)sel;
    state[b * 2 + 1] = kcur - acc;
  }
}

// ---------------------------------------------------------------------------
// compact: streaming scan; append every element with ord >= T as a packed
// 64-bit key (ord<<32)|~idx so one descending sort yields topk tie order.
// ---------------------------------------------------------------------------
__global__ void __launch_bounds__(256)
compact_kernel(const float* __restrict__ preds,
               const unsigned* __restrict__ state,
               unsigned* __restrict__ gcnt,
               unsigned long long* __restrict__ cand) {
  __shared__ __align__(16) float stage[8][2][TILE_ELEMS];
  const int b = blockIdx.y;
  const int tid = threadIdx.x;
  const int wave = tid >> 5, lane = tid & 31;

  const unsigned long long gbase =
      (unsigned long long)(preds + (size_t)b * 3 * HW +
                           (size_t)blockIdx.x * BLK_ELEMS);
  const unsigned T = state[b * 2 + 0];   // exact bit pattern of K-th largest
  const unsigned waveByte = (unsigned)(wave * WAVE_ELEMS * 4);
  const unsigned e0 = (unsigned)(blockIdx.x * BLK_ELEMS + wave * WAVE_ELEMS);

  issue_tile(gbase, waveByte, stage[wave][0], lane);
  int parity = 0;
  for (int t = 0; t < NTILES; ++t) {
    if (t + 1 < NTILES) {
      issue_tile(gbase, waveByte + (unsigned)((t + 1) * TILE_ELEMS * 4),
                 stage[wave][parity ^ 1], lane);
      wait_async_le4();
    } else {
      wait_async_0();
    }
    const float* buf = stage[wave][parity];
#pragma unroll
    for (int s = 0; s < 4; ++s) {
      const float4 v = *(const float4*)(buf + s * 128 + lane * 4);
      float vals[4] = {v.x, v.y, v.z, v.w};
#pragma unroll
      for (int j = 0; j < 4; ++j) {
        unsigned ord = ordOf(vals[j]);
        if (ord >= T) {
          unsigned idx = e0 + (unsigned)(t * TILE_ELEMS + s * 128 + lane * 4 + j);
          unsigned pos = atomicAdd(&gcnt[b], 1u);
          if (pos < CAP)
            cand[(size_t)b * CAP + pos] =
                ((unsigned long long)ord << 32) | (unsigned)(~idx);
        }
      }
    }
    parity ^= 1;
  }
}

// ---------------------------------------------------------------------------
// sort + decode: per batch, bitonic-sort <=2048 candidate keys in LDS
// (ascending; read from the tail for descending), then decode boxes/scores.
// ---------------------------------------------------------------------------
__global__ void __launch_bounds__(1024)
sort_decode_kernel(const float* __restrict__ preds,
                   const unsigned* __restrict__ gcnt,
                   const unsigned long long* __restrict__ cand,
                   float* __restrict__ out_boxes,
                   float* __restrict__ out_scores,
                   float* __restrict__ out_keep) {
  __shared__ unsigned long long keys[CAP];
  const int b = blockIdx.x, tid = threadIdx.x;
  unsigned n = gcnt[b];
  if (n > CAP) n = CAP;
  for (int i = tid; i < CAP; i += 1024)
    keys[i] = (i < (int)n) ? cand[(size_t)b * CAP + i] : 0ull;  // 0 < any real key
  __syncthreads();

  for (unsigned k = 2; k <= CAP; k <<= 1) {
    for (unsigned j = k >> 1; j > 0; j >>= 1) {
      for (unsigned i = tid; i < CAP; i += 1024) {
        unsigned ixj = i ^ j;
        if (ixj > i) {
          unsigned long long a = keys[i], c2 = keys[ixj];
          bool asc = ((i & k) == 0);
          if (asc ? (a > c2) : (a < c2)) { keys[i] = c2; keys[ixj] = a; }
        }
      }
      __syncthreads();
    }
  }

  if (tid < K_TOP) {
    unsigned long long key = keys[CAP - 1 - tid];   // rank tid (desc score, asc idx)
    unsigned ord = (unsigned)(key >> 32);
    unsigned idx = ~(unsigned)key;
    unsigned fb = (ord & 0x80000000u) ? (ord ^ 0x80000000u) : ~ord;
    float score = __uint_as_float(fb);
    float cx = (float)(idx % WIDTH);
    float cy = (float)(idx / WIDTH);
    const float* pb = preds + (size_t)b * 3 * HW;
    float h = fmaxf(pb[HW + idx], 1e-6f) * 640.0f;
    float w = fmaxf(pb[2 * HW + idx], 1e-6f) * 640.0f;
    float* bx = out_boxes + ((size_t)b * K_TOP + tid) * 4;
    bx[0] = cx - 0.5f * w;
    bx[1] = cy - 0.5f * h;
    bx[2] = cx + 0.5f * w;
    bx[3] = cy + 0.5f * h;
    out_scores[(size_t)b * K_TOP + tid] = score;
    out_keep[(size_t)b * K_TOP + tid] = 1.0f;   // keep is identically True
  }
}

// ---------------------------------------------------------------------------
extern "C" void kernel_launch(void* const* d_in, const int* in_sizes, int n_in,
                              void* d_out, int out_size, void* d_ws, size_t ws_size,
                              hipStream_t stream) {
  const float* preds = (const float*)d_in[0];
  float* out = (float*)d_out;
  float* out_boxes  = out;                              // [32,1000,4]
  float* out_scores = out + (size_t)BATCH * K_TOP * 4;  // [32,1000]
  float* out_keep   = out + (size_t)BATCH * K_TOP * 5;  // [32,1000]

  // workspace carve-up (~560 KB)
  unsigned* ghist = (unsigned*)d_ws;           // 32*256 u32
  unsigned* state = ghist + BATCH * 256;       // 32*2   u32  {prefix, kcur}
  unsigned* gcnt  = state + BATCH * 2;         // 32     u32
  unsigned long long* cand =
      (unsigned long long*)(gcnt + BATCH);     // 32*2048 u64 (offset 33152 B, 8-aligned)

  init_kernel<<<dim3((BATCH * 256 + 255) / 256), dim3(256), 0, stream>>>(
      ghist, state, gcnt);

  for (int lvl = 0; lvl < 4; ++lvl) {
    hist_kernel<<<dim3(NBLK, BATCH), dim3(256), 0, stream>>>(preds, ghist,
                                                             state, lvl);
    select_kernel<<<dim3(BATCH), dim3(256), 0, stream>>>(ghist, state);
  }

  compact_kernel<<<dim3(NBLK, BATCH), dim3(256), 0, stream>>>(preds, state,
                                                              gcnt, cand);

  sort_decode_kernel<<<dim3(BATCH), dim3(1024), 0, stream>>>(
      preds, gcnt, cand, out_boxes, out_scores, out_keep);
}